// QuantizedEmbedding_5446018531483
// MI455X (gfx1250) — compile-verified
//
#include <hip/hip_runtime.h>
#include <hip/hip_bf16.h>

// QuantizedEmbedding gather for MI455X (gfx1250, wave32).
//   out[row, :] = fake_quant_lastdim(weight[idx[row], :])
// Memory-bound: ~25 MB traffic -> ~1.1 us floor at 23.3 TB/s.
//
// CDNA5 path: per-wave double-buffered pipeline with
// GLOBAL_LOAD_ASYNC_TO_LDS_B128 (ASYNCcnt) — issue the next row's gather into
// LDS while quantizing the current row. Wave32 absmax reduction done with
// ds_swizzle_b32 XOR-butterflies (1 DS op per step). Output stored with
// non-temporal b128 so the 154MB table stays resident in the 192MB L2.

typedef float f32x4 __attribute__((ext_vector_type(4)));
typedef int   i32x4 __attribute__((ext_vector_type(4)));

#define DIM       768
#define NVEC      6              // f32x4 chunks per lane: 768 / 4 / 32 lanes
#define QMAX_F    127.0f
#define WAVES_PB  8              // 256-thread blocks

#if __has_builtin(__builtin_amdgcn_global_load_async_to_lds_b128) && \
    __has_builtin(__builtin_amdgcn_s_wait_asynccnt)
#define USE_ASYNC_LDS 1
#else
#define USE_ASYNC_LDS 0
#endif

// wave32 XOR-butterfly max step via ds_swizzle_b32 (and_mask=0x1f, or=0, xor=XM)
#define SWIZ_MAX(m, XM)                                                       \
    m = fmaxf(m, __uint_as_float((unsigned)__builtin_amdgcn_ds_swizzle(       \
                     (int)__float_as_uint(m), (((XM) << 10) | 0x1f))))

__device__ __forceinline__ float wave32_absmax(float m) {
    SWIZ_MAX(m, 16);
    SWIZ_MAX(m, 8);
    SWIZ_MAX(m, 4);
    SWIZ_MAX(m, 2);
    SWIZ_MAX(m, 1);
    return m;
}

__device__ __forceinline__ void quant_store_row(const f32x4* __restrict__ v,
                                                float* __restrict__ out,
                                                size_t row, int lane) {
    float m = 0.0f;
    #pragma unroll
    for (int k = 0; k < NVEC; ++k) {
        #pragma unroll
        for (int j = 0; j < 4; ++j)
            m = fmaxf(m, fabsf(v[k][j]));
    }
    m = wave32_absmax(m);

    const float am    = fmaxf(m, 1e-8f);
    const float scale = am * (1.0f / QMAX_F);
    const float inv   = QMAX_F / am;

    f32x4* orow = (f32x4*)(out + row * DIM);
    #pragma unroll
    for (int k = 0; k < NVEC; ++k) {
        f32x4 q;
        #pragma unroll
        for (int j = 0; j < 4; ++j) {
            float r = rintf(v[k][j] * inv);            // RNE, matches jnp.round
            r = fminf(fmaxf(r, -QMAX_F), QMAX_F);      // clip to [-127,127]
            q[j] = r * scale;
        }
        __builtin_nontemporal_store(q, &orow[lane + 32 * k]);
    }
}

#if USE_ASYNC_LDS

// Builtin signature (from hipcc diagnostic): param is a pointer to
// int __vector(4) — cast to typed v4i pointers with explicit address spaces.
typedef __attribute__((address_space(1))) i32x4 gv4i;
typedef __attribute__((address_space(3))) i32x4 lv4i;

#define CAST_G(p) ((gv4i*)(p))
#define CAST_L(p) ((lv4i*)(p))

// Issue one full row (3072B) as 6 async b128 chunk-loads; INST_OFFSET is added
// to BOTH the global and LDS addresses (ISA 08 §4.4), so one base pair works.
#define ASYNC_ROW_LOAD(gbase, lbase)                                              \
    do {                                                                          \
        __builtin_amdgcn_global_load_async_to_lds_b128(CAST_G(gbase),             \
                                                       CAST_L(lbase), 0, 0);      \
        __builtin_amdgcn_global_load_async_to_lds_b128(CAST_G(gbase),             \
                                                       CAST_L(lbase), 512, 0);    \
        __builtin_amdgcn_global_load_async_to_lds_b128(CAST_G(gbase),             \
                                                       CAST_L(lbase), 1024, 0);   \
        __builtin_amdgcn_global_load_async_to_lds_b128(CAST_G(gbase),             \
                                                       CAST_L(lbase), 1536, 0);   \
        __builtin_amdgcn_global_load_async_to_lds_b128(CAST_G(gbase),             \
                                                       CAST_L(lbase), 2048, 0);   \
        __builtin_amdgcn_global_load_async_to_lds_b128(CAST_G(gbase),             \
                                                       CAST_L(lbase), 2560, 0);   \
    } while (0)

__global__ __launch_bounds__(256) void
quant_embed_gather(const float* __restrict__ weight,
                   const int*   __restrict__ indices,
                   float*       __restrict__ out,
                   int n_rows)
{
    // Per-wave LDS double buffer: 2 x 3KB, 8 waves -> 48KB per block.
    __shared__ __align__(16) float smem[WAVES_PB][2][DIM];

    const int lane   = threadIdx.x & 31;
    const int wave   = threadIdx.x >> 5;
    const int stride = (int)gridDim.x * WAVES_PB;

    int row = (int)blockIdx.x * WAVES_PB + wave;
    int cur = 0;

    // Prologue: kick off the async gather of this wave's first row.
    if (row < n_rows) {
        const char* g0 = (const char*)(weight + (size_t)indices[row] * DIM)
                         + lane * 16;
        char* l0 = (char*)smem[wave][0] + lane * 16;
        ASYNC_ROW_LOAD(g0, l0);
    }

    for (; row < n_rows; row += stride) {
        const int  nrow     = row + stride;
        const bool has_next = (nrow < n_rows);

        // Issue next row's gather into the other buffer (overlaps compute).
        if (has_next) {
            const char* gn = (const char*)(weight + (size_t)indices[nrow] * DIM)
                             + lane * 16;
            char* ln = (char*)smem[wave][cur ^ 1] + lane * 16;
            ASYNC_ROW_LOAD(gn, ln);
            __builtin_amdgcn_s_wait_asynccnt(6);  // 6 newest may still be in flight
        } else {
            __builtin_amdgcn_s_wait_asynccnt(0);  // drain: current buffer ready
        }

        // Consume current buffer from LDS (ds_load_b128) and quantize.
        const f32x4* lrow = (const f32x4*)smem[wave][cur];
        f32x4 v[NVEC];
        #pragma unroll
        for (int k = 0; k < NVEC; ++k)
            v[k] = lrow[lane + 32 * k];

        quant_store_row(v, out, (size_t)row, lane);
        cur ^= 1;
    }
}

#else  // fallback: direct b128 gather + global_prefetch_b8 pipelining

__global__ __launch_bounds__(256) void
quant_embed_gather(const float* __restrict__ weight,
                   const int*   __restrict__ indices,
                   float*       __restrict__ out,
                   int n_rows)
{
    const int lane   = threadIdx.x & 31;
    const int wave   = threadIdx.x >> 5;
    const int stride = (int)gridDim.x * WAVES_PB;

    for (int row = (int)blockIdx.x * WAVES_PB + wave; row < n_rows;
         row += stride) {
        const int nrow = row + stride;
        if (nrow < n_rows) {
            const float* np =
                weight + (size_t)indices[nrow] * DIM + lane * (DIM / 32);
            __builtin_prefetch(np, 0, 3);          // global_prefetch_b8
        }

        const f32x4* wrow = (const f32x4*)(weight + (size_t)indices[row] * DIM);
        f32x4 v[NVEC];
        #pragma unroll
        for (int k = 0; k < NVEC; ++k)
            v[k] = wrow[lane + 32 * k];

        quant_store_row(v, out, (size_t)row, lane);
    }
}

#endif  // USE_ASYNC_LDS

extern "C" void kernel_launch(void* const* d_in, const int* in_sizes, int n_in,
                              void* d_out, int out_size, void* d_ws, size_t ws_size,
                              hipStream_t stream)
{
    // setup_inputs() order: weight [VOCAB*DIM] fp32, input [8*512] int
    const float* weight  = (const float*)d_in[0];
    const int*   indices = (const int*)d_in[1];
    float*       out     = (float*)d_out;

    const int n_rows = in_sizes[1];   // 4096 token rows

    const int block = 256;            // 8 wave32s per block
    // 2 rows per wave: one buffer in flight while the other is consumed.
    int grid = (n_rows + 2 * WAVES_PB - 1) / (2 * WAVES_PB);
    if (grid < 1) grid = 1;

    quant_embed_gather<<<grid, block, 0, stream>>>(weight, indices, out, n_rows);
}